// PatchFocalLoss_29523605192774
// MI455X (gfx1250) — compile-verified
//
#include <hip/hip_runtime.h>
#include <hip/hip_bf16.h>

#define BS 512
#define NCOL 8192
#define CHUNK_ELEMS (BS * 2)          // 1024 floats per chunk (b64 per lane)
#define NCHUNK (NCOL / CHUNK_ELEMS)   // 8

#if defined(__HIP_DEVICE_COMPILE__) && defined(__gfx1250__)
#  if __has_builtin(__builtin_amdgcn_global_load_async_to_lds_b64) && \
      __has_builtin(__builtin_amdgcn_s_wait_asynccnt)
#    define HAVE_ASYNC 1
#  else
#    define HAVE_ASYNC 0
#  endif
#else
#  define HAVE_ASYNC 0
#endif

// 64-bit async-copy payload type: pointer-to-v2i32 in explicit target
// address spaces (global = 1, LDS = 3), matching the builtin's signature.
typedef int v2i __attribute__((ext_vector_type(2)));
typedef __attribute__((address_space(1))) v2i* g_v2i_p;
typedef __attribute__((address_space(3))) v2i* l_v2i_p;

// Numerically-stable sigmoid focal loss for one element (matches reference).
__device__ __forceinline__ float focal_elem(float l, float lab) {
  float t     = fmaxf(lab, 0.0f);
  float absl  = fabsf(l);
  float e     = __expf(-absl);                 // exp(-|l|), in (0,1]
  float lp    = __logf(1.0f + e);              // log1p(exp(-|l|))
  float bce   = fmaxf(l, 0.0f) - l * t + lp;
  float inv   = __fdividef(1.0f, 1.0f + e);
  float p     = (l >= 0.0f) ? inv : (e * inv); // sigmoid(l)
  float pt    = t * p + (1.0f - t) * (1.0f - p);
  float om    = 1.0f - pt;
  float focal = om * om;                       // GAMMA = 2
  float aw    = t * 0.75f + (1.0f - t) * 0.25f;
  float loss  = aw * focal * bce;
  return (lab >= 0.0f) ? loss : 0.0f;          // ignore label == -1
}

__global__ __launch_bounds__(BS)
void focal_row_kernel(const float* __restrict__ logits,
                      const float* __restrict__ labels,
                      float* __restrict__ row_sum,
                      unsigned* __restrict__ row_valid) {
  const int tid = threadIdx.x;
  const int row = blockIdx.x;
  const float* grow = logits + (size_t)row * NCOL;
  const float* lrow = labels + (size_t)row * NCOL;

  __shared__ unsigned keys[NCOL];   // neg-loss fp32 bit keys (0 for non-neg)
  __shared__ unsigned hist[256];
  __shared__ float    sred[BS];
  __shared__ unsigned ured[BS];
  __shared__ unsigned sh_prefix;
  __shared__ int      sh_rem;

  unsigned npos_t = 0, posval_t = 0, negkey_t = 0;
  float possum_t = 0.0f;

#if HAVE_ASYNC
  // Double-buffered async global -> LDS staging; each lane owns its slot,
  // ordering via ASYNCcnt only (async loads complete in order).
  __shared__ __align__(16) float sLog[2][CHUNK_ELEMS];
  __shared__ __align__(16) float sLab[2][CHUNK_ELEMS];
  {
    int base = tid * 2;
    __builtin_amdgcn_global_load_async_to_lds_b64(
        (g_v2i_p)(grow + base), (l_v2i_p)(&sLog[0][tid * 2]), 0, 0);
    __builtin_amdgcn_global_load_async_to_lds_b64(
        (g_v2i_p)(lrow + base), (l_v2i_p)(&sLab[0][tid * 2]), 0, 0);
  }
  for (int ch = 0; ch < NCHUNK; ++ch) {
    int buf = ch & 1;
    if (ch + 1 < NCHUNK) {
      int base = (ch + 1) * CHUNK_ELEMS + tid * 2;
      __builtin_amdgcn_global_load_async_to_lds_b64(
          (g_v2i_p)(grow + base), (l_v2i_p)(&sLog[buf ^ 1][tid * 2]), 0, 0);
      __builtin_amdgcn_global_load_async_to_lds_b64(
          (g_v2i_p)(lrow + base), (l_v2i_p)(&sLab[buf ^ 1][tid * 2]), 0, 0);
      __builtin_amdgcn_s_wait_asynccnt(2);   // chunk ch's two loads are done
    } else {
      __builtin_amdgcn_s_wait_asynccnt(0);
    }
#pragma unroll
    for (int j = 0; j < 2; ++j) {
      int   c    = ch * CHUNK_ELEMS + tid * 2 + j;
      float l    = sLog[buf][tid * 2 + j];
      float lab  = sLab[buf][tid * 2 + j];
      float loss = focal_elem(l, lab);
      bool  neg  = (lab == 0.0f);
      keys[c] = neg ? __float_as_uint(loss) : 0u;
      if (lab == 1.0f) { npos_t++; possum_t += loss; posval_t += (loss > 0.0f); }
      negkey_t += (neg && loss > 0.0f) ? 1u : 0u;
    }
  }
#else
  for (int c = tid; c < NCOL; c += BS) {
    __builtin_prefetch(grow + c + 4 * BS, 0, 1);
    __builtin_prefetch(lrow + c + 4 * BS, 0, 1);
    float l    = grow[c];
    float lab  = lrow[c];
    float loss = focal_elem(l, lab);
    bool  neg  = (lab == 0.0f);
    keys[c] = neg ? __float_as_uint(loss) : 0u;
    if (lab == 1.0f) { npos_t++; possum_t += loss; posval_t += (loss > 0.0f); }
    negkey_t += (neg && loss > 0.0f) ? 1u : 0u;
  }
#endif
  __syncthreads();

  // ---- deterministic block reductions ----
  sred[tid] = possum_t; __syncthreads();
  for (int s = BS / 2; s > 0; s >>= 1) { if (tid < s) sred[tid] += sred[tid + s]; __syncthreads(); }
  float possum = sred[0]; __syncthreads();

  ured[tid] = npos_t; __syncthreads();
  for (int s = BS / 2; s > 0; s >>= 1) { if (tid < s) ured[tid] += ured[tid + s]; __syncthreads(); }
  unsigned npos = ured[0]; __syncthreads();

  ured[tid] = posval_t; __syncthreads();
  for (int s = BS / 2; s > 0; s >>= 1) { if (tid < s) ured[tid] += ured[tid + s]; __syncthreads(); }
  unsigned posval = ured[0]; __syncthreads();

  ured[tid] = negkey_t; __syncthreads();
  for (int s = BS / 2; s > 0; s >>= 1) { if (tid < s) ured[tid] += ured[tid + s]; __syncthreads(); }
  unsigned negkeys = ured[0]; __syncthreads();

  // k = n_pos>0 ? max(1, int(n_pos*3.0)) : 10   (exact in fp32 for n_pos<=8192)
  int k = (npos > 0) ? (int)((float)npos * 3.0f) : 10;
  if (npos > 0 && k < 1) k = 1;

  float negsum; unsigned validneg;
  if (k >= (int)negkeys) {
    // keep every negative with loss > 0
    float ps = 0.0f;
    for (int c = tid; c < NCOL; c += BS) ps += __uint_as_float(keys[c]);
    sred[tid] = ps; __syncthreads();
    for (int s = BS / 2; s > 0; s >>= 1) { if (tid < s) sred[tid] += sred[tid + s]; __syncthreads(); }
    negsum = sred[0]; validneg = negkeys;
  } else {
    // MSB-first 8-bit radix select of the k-th largest key (loss>=0 => bits
    // order == numeric order). Integer histograms => deterministic.
    unsigned prefix = 0; int rem = k;
    for (int rnd = 0; rnd < 4; ++rnd) {
      int shift = 24 - 8 * rnd;
      for (int b = tid; b < 256; b += BS) hist[b] = 0;
      __syncthreads();
      for (int c = tid; c < NCOL; c += BS) {
        unsigned key = keys[c];
        bool part = (rnd == 0) || ((key >> (shift + 8)) == prefix);
        if (part) atomicAdd(&hist[(key >> shift) & 255u], 1u);
      }
      __syncthreads();
      if (tid == 0) {
        int r = rem; int sel = 0;
        for (int b = 255; b >= 0; --b) {
          int h = (int)hist[b];
          if (h >= r) { sel = b; break; }
          r -= h;
        }
        sh_prefix = (prefix << 8) | (unsigned)sel;
        sh_rem = r;
      }
      __syncthreads();
      prefix = sh_prefix; rem = sh_rem;
      __syncthreads();
    }
    unsigned T = prefix;                // k-th largest key; T > 0 here
    float ps = 0.0f;
    for (int c = tid; c < NCOL; c += BS) {
      unsigned key = keys[c];
      if (key > T) ps += __uint_as_float(key);
    }
    sred[tid] = ps; __syncthreads();
    for (int s = BS / 2; s > 0; s >>= 1) { if (tid < s) sred[tid] += sred[tid + s]; __syncthreads(); }
    negsum = sred[0] + (float)rem * __uint_as_float(T);  // ties at threshold
    validneg = (unsigned)k;
  }

  if (tid == 0) {
    row_sum[row]   = possum + negsum;
    row_valid[row] = posval + validneg;
  }
}

__global__ __launch_bounds__(BS)
void focal_finalize_kernel(const float* __restrict__ rs,
                           const unsigned* __restrict__ rv,
                           float* __restrict__ out, int B) {
  __shared__ float    s[BS];
  __shared__ unsigned u[BS];
  int t = threadIdx.x;
  float fs = 0.0f; unsigned uv = 0;
  for (int i = t; i < B; i += BS) { fs += rs[i]; uv += rv[i]; }
  s[t] = fs; u[t] = uv; __syncthreads();
  for (int st = BS / 2; st > 0; st >>= 1) {
    if (t < st) { s[t] += s[t + st]; u[t] += u[t + st]; }
    __syncthreads();
  }
  if (t == 0) {
    unsigned nv = u[0];
    out[0] = s[0] / (float)((nv > 0) ? nv : 1u);
  }
}

extern "C" void kernel_launch(void* const* d_in, const int* in_sizes, int n_in,
                              void* d_out, int out_size, void* d_ws, size_t ws_size,
                              hipStream_t stream) {
  const float* logits = (const float*)d_in[0];
  const float* labels = (const float*)d_in[1];
  int B = in_sizes[0] / NCOL;                // 1024 for the reference shapes

  float*    row_sum   = (float*)d_ws;
  unsigned* row_valid = (unsigned*)((char*)d_ws + (size_t)B * sizeof(float));

  focal_row_kernel<<<B, BS, 0, stream>>>(logits, labels, row_sum, row_valid);
  focal_finalize_kernel<<<1, BS, 0, stream>>>(row_sum, row_valid, (float*)d_out, B);
}